// OptimizedTopMAttention_40785009443178
// MI455X (gfx1250) — compile-verified
//
#include <hip/hip_runtime.h>
#include <hip/hip_bf16.h>
#include <math.h>

// Problem constants (from the reference)
#define B_  2
#define N_  4096
#define C_  512
#define C3  1536
#define NH  8
#define HD  64
#define TOPM 128

typedef __attribute__((ext_vector_type(2))) float v2f;
typedef __attribute__((ext_vector_type(8))) float v8f;

// Monotonic float->uint key: larger key <=> larger float (handles negatives).
__device__ __forceinline__ unsigned fkey(float f) {
    unsigned u = __float_as_uint(f);
    return (u & 0x80000000u) ? ~u : (u | 0x80000000u);
}

// ---------------------------------------------------------------------------
// Dense GEMM with bias: C[M,Nout] = A[M,K] @ W[K,Nout] + bias[Nout]
// One 16x16 output tile per wave, K stepped by 4 via v_wmma_f32_16x16x4_f32.
// A-matrix lane layout (ISA 7.12.2, 32-bit 16x4): lanes 0-15 row=lane, K=0,1;
// lanes 16-31 row=lane-16, K=2,3. B symmetric. C/D: VGPR r -> row r + 8*(lane>=16).
// ---------------------------------------------------------------------------
__global__ __launch_bounds__(256) void wmma_gemm_bias(
    const float* __restrict__ A, const float* __restrict__ W,
    const float* __restrict__ bias, float* __restrict__ Cout,
    int M, int Nout, int K)
{
    int wave  = (int)((blockIdx.x * blockDim.x + threadIdx.x) >> 5);
    int lane  = threadIdx.x & 31;
    int tilesN = Nout >> 4;
    int tileM = wave / tilesN;
    int tileN = wave - tileM * tilesN;
    if (tileM * 16 >= M) return;

    int lr = lane & 15;              // row-in-tile (A) / col-in-tile (B)
    int kb = (lane >> 4) << 1;       // K sub-offset: 0 or 2

    int rowA = tileM * 16 + lr;
    int colB = tileN * 16 + lr;
    const float* arow = A + (size_t)rowA * K;

    v8f acc = {};
    for (int kk = 0; kk < K; kk += 4) {
        v2f a, b;
        a.x = arow[kk + kb];
        a.y = arow[kk + kb + 1];
        b.x = W[(size_t)(kk + kb)     * Nout + colB];
        b.y = W[(size_t)(kk + kb + 1) * Nout + colB];
        acc = __builtin_amdgcn_wmma_f32_16x16x4_f32(
                  false, a, false, b, (short)0, acc, false, false);
    }

    float bv = bias[colB];
    int rbase = (lane >> 4) << 3;
#pragma unroll
    for (int r = 0; r < 8; ++r) {
        Cout[(size_t)(tileM * 16 + rbase + r) * Nout + colB] = acc[r] + bv;
    }
}

// ---------------------------------------------------------------------------
// Fused top-M attention. One workgroup = 16 query rows of one (b,h).
// 16 waves. Score slab S[16][4096] lives entirely in LDS (CDNA5: 320KB/WG).
//   Phase A: stage Q tile (16x64) in LDS.
//   Phase B: S = scale * Q K^T via wmma f32 16x16x4 (each wave: 16 col-tiles).
//   Phase C: per-wave exact 128th-largest via 4-pass radix select on fkey,
//            histograms in LDS via ds_add atomics (wave-internal, in-order).
//   Phase D: keep >T plus lowest-index ties ==T (ballot prefix counts) ->
//            exactly 128 kept, matching jax top_k; exp/sum for softmax.
//   Phase E: sparse PV: 128 gathered V rows, 2 dims/lane.
// ---------------------------------------------------------------------------
#define QSTRIDE 65     // 64 + 1 pad: conflict-free column reads of Q
#define SSTRIDE 4100   // 4096 + 4 pad: de-conflict row+8 vs row writes

__global__ __launch_bounds__(512) void topm_attn(
    const float* __restrict__ qkv, float* __restrict__ att)
{
    extern __shared__ unsigned char smem_raw[];
    float*    q_s  = (float*)smem_raw;                 // 16*QSTRIDE floats
    float*    S_s  = q_s + 16 * QSTRIDE;               // 16*SSTRIDE floats
    unsigned* hist = (unsigned*)(S_s + 16 * SSTRIDE);  // 16*256 u32
    int*      kidx = (int*)(hist + 16 * 256);          // 16*128 int
    float*    kval = (float*)(kidx + 16 * TOPM);       // 16*128 float

    int blk = blockIdx.x;                // B*NH*(N/16)
    int rt  = blk % (N_ / 16);
    int bh  = blk / (N_ / 16);
    int h   = bh % NH;
    int b   = bh / NH;

    int tid  = threadIdx.x;
    int lane = tid & 31;
    int wave = tid >> 5;                 // 0..15 == row within tile
    int row0 = rt * 16;

    // ---- Phase A: load Q tile ----
    for (int i = tid; i < 16 * HD; i += 512) {
        int r = i >> 6, d = i & 63;
        q_s[r * QSTRIDE + d] = qkv[((size_t)(b * N_ + row0 + r) * C3) + h * HD + d];
    }
    __syncthreads();

    const float scale = 0.125f;          // hd^-0.5

    // ---- Phase B: S = scale * Q K^T ----
    {
        int lr = lane & 15;
        int kb = (lane >> 4) << 1;
        int rbase = (lane >> 4) << 3;
        for (int ct = wave; ct < N_ / 16; ct += 16) {
            int ncol = ct * 16 + lr;
            const float* krow = qkv + ((size_t)(b * N_ + ncol) * C3) + C_ + h * HD;
            v8f acc = {};
#pragma unroll
            for (int kk = 0; kk < HD; kk += 4) {
                v2f a, bb;
                a.x  = q_s[lr * QSTRIDE + kk + kb];
                a.y  = q_s[lr * QSTRIDE + kk + kb + 1];
                bb.x = krow[kk + kb];
                bb.y = krow[kk + kb + 1];
                acc = __builtin_amdgcn_wmma_f32_16x16x4_f32(
                          false, a, false, bb, (short)0, acc, false, false);
            }
#pragma unroll
            for (int r = 0; r < 8; ++r)
                S_s[(rbase + r) * SSTRIDE + ct * 16 + lr] = acc[r] * scale;
        }
    }
    __syncthreads();

    // ---- Phase C: radix-select 128th-largest key (per wave / per row) ----
    const float* Srow  = S_s + wave * SSTRIDE;
    unsigned*    myhist = hist + wave * 256;
    unsigned prefix = 0, maskb = 0;
    int kneed = TOPM;
    for (int pass = 0; pass < 4; ++pass) {
        int shift = 24 - pass * 8;
        for (int i = lane; i < 256; i += 32) myhist[i] = 0u;
        for (int i = lane; i < N_; i += 32) {
            unsigned key = fkey(Srow[i]);
            if ((key & maskb) == prefix)
                atomicAdd(&myhist[(key >> shift) & 255u], 1u);
        }
        // all lanes scan identically (no divergence): descending bins
        unsigned cum = 0; int chosen = 0;
        for (int bin = 255; bin >= 0; --bin) {
            unsigned hc = myhist[bin];
            if (cum + hc >= (unsigned)kneed) { chosen = bin; break; }
            cum += hc;
        }
        kneed  -= (int)cum;
        prefix |= ((unsigned)chosen) << shift;
        maskb  |= 0xFFu << shift;
    }
    const unsigned T   = prefix;   // key of the 128th largest
    const int      rEq = kneed;    // # of ==T entries to keep (lowest index first)

    // ---- Phase D: build kept list (exactly 128), softmax stats ----
    float m = -INFINITY;
    for (int i = lane; i < N_; i += 32) m = fmaxf(m, Srow[i]);
#pragma unroll
    for (int off = 16; off > 0; off >>= 1)
        m = fmaxf(m, __shfl_xor(m, off, 32));

    int*   myidx = kidx + wave * TOPM;
    float* myval = kval + wave * TOPM;
    unsigned lmask_lt = (1u << lane) - 1u;
    int eqSeen = 0, kept = 0;
    float sumE = 0.f;
    for (int base = 0; base < N_; base += 32) {
        int i = base + lane;
        float s = Srow[i];
        unsigned key = fkey(s);
        bool isG = key > T;
        bool isE = (key == T);
        unsigned ebal = (unsigned)__ballot(isE);
        bool keep = isG || (isE && (eqSeen + __popc(ebal & lmask_lt) < rEq));
        unsigned kbal = (unsigned)__ballot(keep);
        if (keep) {
            float e = __expf(s - m);
            int pos = kept + __popc(kbal & lmask_lt);
            myidx[pos] = i;
            myval[pos] = e;
            sumE += e;
        }
        kept   += __popc(kbal);
        eqSeen += __popc(ebal);
    }
#pragma unroll
    for (int off = 16; off > 0; off >>= 1)
        sumE += __shfl_xor(sumE, off, 32);
    float inv = 1.f / sumE;

    // ---- Phase E: sparse PV (128 rows), 2 output dims per lane ----
    float acc0 = 0.f, acc1 = 0.f;
    int d0 = lane, d1 = lane + 32;
    for (int j = 0; j < TOPM; ++j) {
        int idx = myidx[j];          // LDS broadcast
        float p = myval[j];
        const float* vrow = qkv + ((size_t)(b * N_ + idx) * C3) + 2 * C_ + h * HD;
        acc0 += p * vrow[d0];
        acc1 += p * vrow[d1];
    }
    int nrow = row0 + wave;
    float* orow = att + ((size_t)(b * N_ + nrow) * C_) + h * HD;
    orow[d0] = acc0 * inv;
    orow[d1] = acc1 * inv;
}

// ---------------------------------------------------------------------------
extern "C" void kernel_launch(void* const* d_in, const int* in_sizes, int n_in,
                              void* d_out, int out_size, void* d_ws, size_t ws_size,
                              hipStream_t stream)
{
    const float* x      = (const float*)d_in[0];
    const float* qkv_w  = (const float*)d_in[1];
    const float* qkv_b  = (const float*)d_in[2];
    const float* proj_w = (const float*)d_in[3];
    const float* proj_b = (const float*)d_in[4];
    float* out = (float*)d_out;

    float* qkv = (float*)d_ws;                         // [B*N, 3C] = 50.3 MB
    float* att = qkv + (size_t)B_ * N_ * C3;           // [B*N, C]  = 16.8 MB

    // K1: qkv = x @ qkv_w + qkv_b   (M=8192, N=1536, K=512)
    {
        int tiles = (B_ * N_ / 16) * (C3 / 16);        // 49152 tiles, 8 waves/blk
        wmma_gemm_bias<<<tiles / 8, 256, 0, stream>>>(x, qkv_w, qkv_b, qkv,
                                                      B_ * N_, C3, C_);
    }
    // K2: fused top-128 attention (LDS-resident 16x4096 score slab)
    {
        size_t shmem = (size_t)(16 * QSTRIDE + 16 * SSTRIDE) * sizeof(float)
                     + (size_t)(16 * 256) * sizeof(unsigned)
                     + (size_t)(16 * TOPM) * (sizeof(int) + sizeof(float));
        topm_attn<<<B_ * NH * (N_ / 16), 512, shmem, stream>>>(qkv, att);
    }
    // K3: out = att @ proj_w + proj_b  (M=8192, N=512, K=512)
    {
        int tiles = (B_ * N_ / 16) * (C_ / 16);        // 16384 tiles
        wmma_gemm_bias<<<tiles / 8, 256, 0, stream>>>(att, proj_w, proj_b, out,
                                                      B_ * N_, C_, C_);
    }
}